// NonLinearReadoutBlock_2370821947753
// MI455X (gfx1250) — compile-verified
//
#include <hip/hip_runtime.h>
#include <hip/hip_bf16.h>
#include <math.h>
#include <stdint.h>

typedef __attribute__((ext_vector_type(2))) float v2f;
typedef __attribute__((ext_vector_type(4))) float v4f;
typedef __attribute__((ext_vector_type(8))) float v8f;
typedef __attribute__((ext_vector_type(4))) unsigned v4u;
typedef __attribute__((ext_vector_type(4))) int v4i;
typedef __attribute__((ext_vector_type(8))) int v8i;

#define MUL_IN   128
#define XCOLS    512
#define XPITCH   516     // padded row pitch (floats): 516 mod 64 == 4 -> conflict-free strided reads
#define WAVES_PB 4
#define THREADS  (WAVES_PB * 32)

// TDM pad config reproduces the 516-pitch: +2 DWORDs after every 256 DWORDs stored.
// LDS column of x element (r, k):  r*516 + k + 2*(k>>8)

// LDS layout (float offsets)
#define OFF_W1SF 0                        // W1_s fragments [kk=32][col=32][hi=2][2] = 4096
#define OFF_W1VF (OFF_W1SF + 4096)       // W1_v fragments [kk=32][col=16][hi=2][2] = 2048
#define OFF_B1   (OFF_W1VF + 2048)       // 32
#define OFF_W2S  (OFF_B1 + 32)           // 160 (16x10)
#define OFF_W2V  (OFF_W2S + 160)         // 16
#define OFF_B2   (OFF_W2V + 16)          // 16 (10 used)
#define OFF_XT   (OFF_B2 + 16)           // per-wave x tiles
#define WTILE    (16 * XPITCH + 8)       // + tail slack for TDM end-of-tile pad
#define LDS_FLOATS (OFF_XT + WAVES_PB * WTILE)

#if defined(__has_builtin)
#if __has_builtin(__builtin_amdgcn_tensor_load_to_lds) && __has_builtin(__builtin_amdgcn_s_wait_tensorcnt)
#define TDM_OK 1
#endif
#endif

__device__ __forceinline__ float silu_f(float v) {
    return v / (1.0f + __expf(-v));
}

__global__ void __launch_bounds__(THREADS)
readout_wmma_kernel(const float* __restrict__ x,
                    const float* __restrict__ W1s,
                    const float* __restrict__ W1v,
                    const float* __restrict__ b1p,
                    const float* __restrict__ W2sp,
                    const float* __restrict__ W2vp,
                    const float* __restrict__ b2p,
                    float* __restrict__ out,
                    int N)
{
    extern __shared__ float lds[];
    const int tid = threadIdx.x;

    // ---- stage layer-1 weights pre-swizzled into WMMA B-fragment order ----
    // w1s_frag[i], i = kk*128 + col*4 + hi*2 + j  holds W1_s[4kk + 2hi + j][col]
    for (int i = tid; i < 4096; i += THREADS) {
        int kk  = i >> 7;
        int col = (i >> 2) & 31;
        int hi2 = (i >> 1) & 1;
        int j   = i & 1;
        lds[OFF_W1SF + i] = W1s[(4 * kk + 2 * hi2 + j) * 32 + col];
    }
    // w1v_frag[i], i = kk*64 + col*4 + hi*2 + j  holds W1_v[4kk + 2hi + j][col]
    for (int i = tid; i < 2048; i += THREADS) {
        int kk  = i >> 6;
        int col = (i >> 2) & 15;
        int hi2 = (i >> 1) & 1;
        int j   = i & 1;
        lds[OFF_W1VF + i] = W1v[(4 * kk + 2 * hi2 + j) * 16 + col];
    }
    if (tid < 32)  lds[OFF_B1  + tid] = b1p[tid];
    for (int i = tid; i < 160; i += THREADS) lds[OFF_W2S + i] = W2sp[i];
    if (tid < 16)  lds[OFF_W2V + tid] = W2vp[tid];
    if (tid < 10)  lds[OFF_B2  + tid] = b2p[tid];
    __syncthreads();

    const int w    = tid >> 5;
    const int lane = tid & 31;
    const int row  = lane & 15;     // column n inside 16x16 tiles
    const int hi   = lane >> 4;     // K-half selector for A/B fragments

    const int numTiles = (N + 15) >> 4;
    const int tile = blockIdx.x * WAVES_PB + w;
    if (tile >= numTiles) return;
    const int base = tile * 16;

    float* xt = lds + OFF_XT + w * WTILE;

#if TDM_OK
    // ---- one TDM instruction stages the whole 16x512 f32 tile, padded in-flight ----
    {
        unsigned lds_byte = (unsigned)(uintptr_t)(void*)xt;
        unsigned long long ga = (unsigned long long)(uintptr_t)(x + (size_t)base * XCOLS);
        unsigned td1 = (unsigned)(N - base);   // rows available; OOB rows read as zero
        v4u g0;
        g0.x = 1u;                              // count=1 valid descriptor
        g0.y = lds_byte;                        // lds_addr
        g0.z = (unsigned)ga;                    // global_addr[31:0]
        g0.w = (unsigned)((ga >> 32) & 0x1FFFFFFu) | (2u << 30);  // addr[56:32] | type=2
        v8i g1;
        g1.s0 = (int)((2u << 16)    // data_size = 4B
                    | (1u << 20)    // pad_enable
                    | (7u << 22)    // pad_interval = 256 DWORDs
                    | (1u << 25));  // pad_amount   = 2 DWORDs
        g1.s1 = (int)(512u << 16);                      // tensor_dim0 = 512 (lo16 at bit48)
        g1.s2 = (int)((td1 & 0xffffu) << 16);           // tensor_dim0 hi=0 | tensor_dim1 lo
        g1.s3 = (int)((td1 >> 16) | (512u << 16));      // tensor_dim1 hi | tile_dim0 = 512
        g1.s4 = 16;                                     // tile_dim1 = 16 rows, tile_dim2 = 0
        g1.s5 = 512;                                    // tensor_dim0_stride = 512
        g1.s6 = 0;
        g1.s7 = 0;
        v4i zz4; zz4.x = 0; zz4.y = 0; zz4.z = 0; zz4.w = 0;
        v8i zz8; zz8.s0 = 0; zz8.s1 = 0; zz8.s2 = 0; zz8.s3 = 0;
                 zz8.s4 = 0; zz8.s5 = 0; zz8.s6 = 0; zz8.s7 = 0;
        __builtin_amdgcn_tensor_load_to_lds(g0, g1, zz4, zz4, zz8, 0);
        __builtin_amdgcn_s_wait_tensorcnt(0);
        asm volatile("" ::: "memory");
    }
#else
    // ---- fallback: coalesced b128 global loads + padded b64 LDS stores ----
    for (int r = 0; r < 16; ++r) {
        int gr = base + r; if (gr >= N) gr = N - 1;
        const v4f* src = (const v4f*)(x + (size_t)gr * XCOLS);
        #pragma unroll
        for (int q = 0; q < 4; ++q) {
            v4f d = src[q * 32 + lane];
            int c0 = (q * 32 + lane) * 4;
            int dofs = r * XPITCH + c0 + 2 * (c0 >> 8);
            v2f lo; lo.x = d.x; lo.y = d.y;
            v2f hip2; hip2.x = d.z; hip2.y = d.w;
            *(v2f*)(xt + dofs)     = lo;
            *(v2f*)(xt + dofs + 2) = hip2;
        }
    }
#endif

    // ---- layer 1: D = X(16x128) @ W(128x{32,16}) via 32 K-steps of 16x16x4 ----
    v8f acc0 = {};      // h_s cols 0..15  (scalars pre-act)
    v8f acc1 = {};      // h_s cols 16..31 (gates pre-act)
    v8f accv[3] = {};   // h_v per component
    const int rowbase = row * XPITCH;

    // Per-lane base pointers: all inner-loop DS accesses become immediate offsets.
    const float* as_base = xt + rowbase + 2 * hi;               // a_s pair at +16*kk bytes
    const float* av_base = xt + rowbase + 6 * hi;               // x_v region base (hi-folded)
    const v2f*   bs_base = (const v2f*)(lds + OFF_W1SF) + row * 2 + hi;
    const v2f*   bv_base = (const v2f*)(lds + OFF_W1VF) + row * 2 + hi;
    const int    hi6 = 6 * hi;

    #pragma unroll
    for (int kk = 0; kk < 32; ++kk) {
        v2f a_s = *(const v2f*)(as_base + 4 * kk);    // aligned b64, const offset
        v2f b0  = bs_base[kk * 64];                   // cols 0..15
        v2f b1f = bs_base[kk * 64 + 32];              // cols 16..31
        v2f bv  = bv_base[kk * 32];

        acc0 = __builtin_amdgcn_wmma_f32_16x16x4_f32(false, a_s, false, b0,
                                                     (short)0, acc0, false, false);
        acc1 = __builtin_amdgcn_wmma_f32_16x16x4_f32(false, a_s, false, b1f,
                                                     (short)0, acc1, false, false);
        #pragma unroll
        for (int c = 0; c < 3; ++c) {
            // column (minus the 6*hi already in av_base): compile-time per (kk,c)
            const int colA = MUL_IN + 12 * kk + c;
            const int colB = colA + 3;
            // pad adjust: +2 once past column 256; constant except at kk==10
            const int adjA = 2 * ((colA + hi6) >> 8);
            const int adjB = 2 * ((colB + hi6) >> 8);
            v2f a_v;
            a_v.x = av_base[colA + adjA];
            a_v.y = av_base[colB + adjB];
            accv[c] = __builtin_amdgcn_wmma_f32_16x16x4_f32(false, a_v, false, bv,
                                                            (short)0, accv[c], false, false);
        }
    }

    // ---- epilogue: SiLU + gate, bounce via wave-private LDS (reuse xt) ----
    const float inv1 = 0.08838834764831845f;   // 1/sqrt(128)
    const float inv2 = 0.25f;                  // 1/sqrt(16)
    float* sc = xt;            // [16][17]
    float* gv = xt + 272;      // [3][16][17]
    const float bs = lds[OFF_B1 + row];
    const float bg = lds[OFF_B1 + 16 + row];

    #pragma unroll
    for (int v = 0; v < 8; ++v) {
        const int m = v + 8 * hi;              // C/D layout: VGPR v -> row m
        float h0 = acc0[v] * inv1 + bs;
        float h1 = acc1[v] * inv1 + bg;
        float s = silu_f(h0);
        float g = silu_f(h1);
        sc[m * 17 + row] = s;
        gv[0 * 272 + m * 17 + row] = g * (accv[0][v] * inv1);
        gv[1 * 272 + m * 17 + row] = g * (accv[1][v] * inv1);
        gv[2 * 272 + m * 17 + row] = g * (accv[2][v] * inv1);
    }
    // same-wave DS ops are in-order: stores above visible to loads below

    const int m  = row;
    const int gr = base + m;
    if (gr < N) {
        float* op = out + (size_t)gr * 13;
        if (hi == 0) {
            // scalar head: out_s = silu(h_s) @ W2_s * inv2 + b2
            float o[10];
            #pragma unroll
            for (int j = 0; j < 10; ++j) o[j] = 0.0f;
            #pragma unroll 4
            for (int k = 0; k < 16; ++k) {
                float s = sc[m * 17 + k];
                #pragma unroll
                for (int j = 0; j < 10; ++j)
                    o[j] += s * lds[OFF_W2S + k * 10 + j];
            }
            #pragma unroll
            for (int j = 0; j < 10; ++j)
                op[j] = o[j] * inv2 + lds[OFF_B2 + j];
        } else {
            // vector head: out_v[c] = (gated_v[:,c] . W2_v) * inv2
            float o0 = 0.f, o1 = 0.f, o2 = 0.f;
            #pragma unroll 4
            for (int k = 0; k < 16; ++k) {
                float wv = lds[OFF_W2V + k];
                o0 += gv[0 * 272 + m * 17 + k] * wv;
                o1 += gv[1 * 272 + m * 17 + k] * wv;
                o2 += gv[2 * 272 + m * 17 + k] * wv;
            }
            op[10] = o0 * inv2;
            op[11] = o1 * inv2;
            op[12] = o2 * inv2;
        }
    }
}

extern "C" void kernel_launch(void* const* d_in, const int* in_sizes, int n_in,
                              void* d_out, int out_size, void* d_ws, size_t ws_size,
                              hipStream_t stream) {
    (void)n_in; (void)out_size; (void)d_ws; (void)ws_size;
    const float* x   = (const float*)d_in[0];
    const float* W1s = (const float*)d_in[1];
    const float* W1v = (const float*)d_in[2];
    const float* b1  = (const float*)d_in[3];
    const float* W2s = (const float*)d_in[4];
    const float* W2v = (const float*)d_in[5];
    const float* b2  = (const float*)d_in[6];
    float* out = (float*)d_out;

    const int N = in_sizes[0] / (MUL_IN * 4);
    const int numTiles = (N + 15) >> 4;
    const int blocks = (numTiles + WAVES_PB - 1) / WAVES_PB;
    const size_t shmem = (size_t)LDS_FLOATS * sizeof(float);

    readout_wmma_kernel<<<blocks, THREADS, shmem, stream>>>(
        x, W1s, W1v, b1, W2s, W2v, b2, out, N);
}